// GatedGCNNet_68496138436961
// MI455X (gfx1250) — compile-verified
//
#include <hip/hip_runtime.h>
#include <math.h>

#define N_NODES 20000
#define M_EDGES 160000
#define IN_DIM  128
#define HID     32
#define BIGC    1152   // 32 emb + 32 p2 + 32 u1 + 32 u2 + 1024 Z
#define L_LAYERS 4

typedef __attribute__((ext_vector_type(2))) float v2f;
typedef __attribute__((ext_vector_type(8))) float v8f;

// ---------------------------------------------------------------------------
// f32 WMMA GEMM with compile-time shapes:
//   C[rows,cols] = A[rows,K] @ B[K,cols] + bias
// One wave per 16x16 output tile; K stepped by 4 via V_WMMA_F32_16X16X4_F32.
// LDA/LDB/LDC/K constant -> fully unrolled loop, immediate-offset loads,
// zero per-iteration address VALU.
// ---------------------------------------------------------------------------
template <int LDA, int LDB, int LDC, int K>
__global__ void wmma_gemm_f32(const float* __restrict__ A,
                              const float* __restrict__ B,
                              const float* __restrict__ bias,
                              float* __restrict__ C,
                              int rowTiles, int colTiles) {
  int gw   = (int)((blockIdx.x * blockDim.x + threadIdx.x) >> 5);
  int lane = threadIdx.x & 31;
  int nT = rowTiles * colTiles;
  if (gw >= nT) return;                 // uniform per-wave -> EXEC all 1s inside
  int rt = gw / colTiles;
  int ct = gw - rt * colTiles;
  int row0 = rt << 4, col0 = ct << 4;
  int lm = lane & 15, hi = lane >> 4;

  // A frag: lane lm holds A[m][k+2*hi], A[m][k+2*hi+1]
  const float* Ap = A + (size_t)(row0 + lm) * LDA + (hi << 1);
  // B frag: lane lm holds B[k+2*hi][n], B[k+2*hi+1][n]
  const float* Bp = B + (size_t)(hi << 1) * LDB + col0 + lm;

  v8f c = {};
#pragma unroll
  for (int k = 0; k < K; k += 4) {
    v2f a, b;
    a.x = Ap[k];                        // contiguous pair -> global_load_b64
    a.y = Ap[k + 1];
    b.x = Bp[(size_t)k * LDB];          // immediate offsets (k*LDB constant)
    b.y = Bp[(size_t)k * LDB + LDB];
    c = __builtin_amdgcn_wmma_f32_16x16x4_f32(false, a, false, b,
                                              (short)0, c, false, false);
  }
  int ocol = col0 + lm;
  float bv = bias ? bias[ocol] : 0.0f;
  int orow = row0 + hi * 8;
#pragma unroll
  for (int r = 0; r < 8; ++r)
    C[(size_t)(orow + r) * LDC + ocol] = c[r] + bv;
}

// ---------------------------------------------------------------------------
// Build fused node weight matrix Wbig[128,1152] + bias[1152].
// Conv1d(2->1,k=3,pad=1) folded into effective 128x32 matrices per channel.
// ---------------------------------------------------------------------------
__global__ void build_wbig(const float* __restrict__ Wemb, const float* __restrict__ bemb,
                           const float* __restrict__ Wp1,  const float* __restrict__ bp1,
                           const float* __restrict__ Wp2,  const float* __restrict__ bp2,
                           const float* __restrict__ convw, const float* __restrict__ convb,
                           const float* __restrict__ Wep2, const float* __restrict__ bep2,
                           float* __restrict__ Wbig, float* __restrict__ bbig) {
  int i = blockIdx.x * blockDim.x + threadIdx.x;
  if (i >= IN_DIM * BIGC) return;
  int x = i / BIGC, col = i - x * BIGC;
  float v;
  if (col < 32) {
    v = Wemb[x * 32 + col];
  } else if (col < 64) {
    v = Wp2[x * 32 + (col - 32)];
  } else if (col < 128) {
    int ch = (col < 96) ? 0 : 1;
    int j = col & 31;
    const float* cw = convw + ch * 3;      // conv_w[0][ch][0..2]
    v = cw[1] * Wep2[x * 32 + j];
    if (x + 1 < IN_DIM) v += cw[0] * Wep2[(x + 1) * 32 + j];
    if (x >= 1)         v += cw[2] * Wep2[(x - 1) * 32 + j];
  } else {
    v = Wp1[(size_t)x * 1024 + (col - 128)];
  }
  Wbig[(size_t)x * BIGC + col] = v;
  if (x == 0) {
    float bv;
    if (col < 32)      bv = bemb[col];
    else if (col < 64) bv = bp2[col - 32];
    else if (col < 96) {                   // conv bias folded here (u1 block)
      int j = col - 64;
      float cs = 0.f;
      for (int t = 0; t < IN_DIM; ++t) cs += Wep2[t * 32 + j];
      bv = bep2[j] + convb[0] * cs;
    } else if (col < 128) bv = 0.f;        // u2 block: bias already in u1
    else bv = bp1[col - 128];
    bbig[col] = bv;
  }
}

// hh[n,j] = Znode[n, j]  (extract embedding block)
__global__ void copy_hh(const float* __restrict__ Z, float* __restrict__ hh, long n) {
  long i = blockIdx.x * (long)blockDim.x + threadIdx.x;
  long stride = gridDim.x * (long)blockDim.x;
  for (; i < n; i += stride)
    hh[i] = Z[(i >> 5) * (long)BIGC + (i & 31)];
}

// ---------------------------------------------------------------------------
// Edge score: one wave per edge.
//  y[a]   = Z[src][a,:] . p2[dst]          (32 MACs/lane)
//  t      = y @ W_ep3 + b_ep3              (shfl reduction)
//  epre   = t + u1[src] + u2[dst]
// ---------------------------------------------------------------------------
__global__ void edge_score(const float* __restrict__ Z,
                           const int* __restrict__ src, const int* __restrict__ dst,
                           const float* __restrict__ Wep3, const float* __restrict__ bep3,
                           float* __restrict__ epre, int M) {
  int lane = threadIdx.x & 31;
  int m = (int)((blockIdx.x * blockDim.x + threadIdx.x) >> 5);
  if (m >= M) return;
  int s = src[m], d = dst[m];
  const float* Zs = Z + (size_t)s * BIGC;
  const float* Zd = Z + (size_t)d * BIGC;
  const float* Zrow = Zs + 128 + lane * 32;
  const float* p2 = Zd + 32;
  float y = 0.f;
#pragma unroll
  for (int k = 0; k < 32; ++k) y += Zrow[k] * p2[k];
  float t = bep3[lane];
#pragma unroll
  for (int a = 0; a < 32; ++a)
    t += __shfl(y, a, 32) * Wep3[a * 32 + lane];
  epre[(size_t)m * 32 + lane] = t + Zs[64 + lane] + Zd[96 + lane];
}

// ---------------------------------------------------------------------------
// Column stats (sum, sum^2) over X[R,32] -> stats[0..31], stats[32..63]
// ---------------------------------------------------------------------------
__global__ void colstats(const float* __restrict__ X, int R, float* __restrict__ stats) {
  int col = threadIdx.x & 31;
  int rg  = threadIdx.x >> 5;             // 0..7
  float s = 0.f, s2 = 0.f;
  for (long r = blockIdx.x * 8 + rg; r < R; r += (long)gridDim.x * 8) {
    float v = X[r * 32 + col];
    s += v; s2 += v * v;
  }
  __shared__ float sh[2][8][32];
  sh[0][rg][col] = s; sh[1][rg][col] = s2;
  __syncthreads();
  if (rg == 0) {
    for (int i = 1; i < 8; ++i) { s += sh[0][i][col]; s2 += sh[1][i][col]; }
    atomicAdd(&stats[col], s);
    atomicAdd(&stats[32 + col], s2);
  }
}

__global__ void bn_finalize(float* __restrict__ stats, float invR) {
  int j = threadIdx.x;
  if (j < 32) {
    float mu = stats[j] * invR;
    float var = stats[32 + j] * invR - mu * mu;
    stats[64 + j] = mu;
    stats[96 + j] = rsqrtf(var + 1e-5f);
  }
}

// out = (accumulate? out : 0) + relu(bn(x))
__global__ void bn_relu_apply(const float* __restrict__ x, const float* __restrict__ stats,
                              const float* __restrict__ g, const float* __restrict__ b,
                              float* __restrict__ out, long n, int accumulate) {
  long i = blockIdx.x * (long)blockDim.x + threadIdx.x;
  long stride = gridDim.x * (long)blockDim.x;
  for (; i < n; i += stride) {
    int j = (int)(i & 31);
    float v = (x[i] - stats[64 + j]) * stats[96 + j] * g[j] + b[j];
    v = v > 0.f ? v : 0.f;
    if (accumulate) out[i] += v; else out[i] = v;
  }
}

__global__ void zero_f(float* __restrict__ p, long n) {
  long i = blockIdx.x * (long)blockDim.x + threadIdx.x;
  long stride = gridDim.x * (long)blockDim.x;
  for (; i < n; i += stride) p[i] = 0.f;
}

// Wcat[32,128] = [A|B|D|E] weights for layer l, bcat[128] biases
__global__ void build_wcat(const float* __restrict__ Aw, const float* __restrict__ Bw,
                           const float* __restrict__ Dw, const float* __restrict__ Ew,
                           const float* __restrict__ Ab, const float* __restrict__ Bb,
                           const float* __restrict__ Db, const float* __restrict__ Eb,
                           int l, float* __restrict__ Wcat, float* __restrict__ bcat) {
  int i = blockIdx.x * blockDim.x + threadIdx.x;
  if (i >= 32 * 128) return;
  int row = i >> 7, col = i & 127;
  int blk = col >> 5, j = col & 31;
  const float* W = (blk == 0) ? Aw : (blk == 1) ? Bw : (blk == 2) ? Dw : Ew;
  Wcat[i] = W[l * 1024 + row * 32 + j];
  if (row == 0) {
    const float* bb = (blk == 0) ? Ab : (blk == 1) ? Bb : (blk == 2) ? Db : Eb;
    bcat[col] = bb[l * 32 + j];
  }
}

// epre += Dh[src]+Eh[dst]; sig=sigmoid(epre); scatter-add num/den
__global__ void edge_update(float* __restrict__ epre, const float* __restrict__ ABDE,
                            const int* __restrict__ src, const int* __restrict__ dst,
                            float* __restrict__ num, float* __restrict__ den, long n) {
  long i = blockIdx.x * (long)blockDim.x + threadIdx.x;
  long stride = gridDim.x * (long)blockDim.x;
  for (; i < n; i += stride) {
    int j = (int)(i & 31);
    long m = i >> 5;
    int s = src[m], d = dst[m];
    float v = epre[i] + ABDE[(size_t)s * 128 + 64 + j] + ABDE[(size_t)d * 128 + 96 + j];
    epre[i] = v;
    float sig = 1.f / (1.f + expf(-v));
    atomicAdd(&num[(size_t)d * 32 + j], sig * ABDE[(size_t)s * 128 + 32 + j]);
    atomicAdd(&den[(size_t)d * 32 + j], sig);
  }
}

// hpre = Ah + num/(den+eps)
__global__ void node_combine(const float* __restrict__ ABDE, const float* __restrict__ num,
                             const float* __restrict__ den, float* __restrict__ hpre, long n) {
  long i = blockIdx.x * (long)blockDim.x + threadIdx.x;
  long stride = gridDim.x * (long)blockDim.x;
  for (; i < n; i += stride) {
    long nd = i >> 5; int j = (int)(i & 31);
    hpre[i] = ABDE[nd * 128 + j] + num[i] / (den[i] + 1e-6f);
  }
}

// mean readout + tiny MLP (720 MACs, single thread)
__global__ void readout(const float* __restrict__ sums,
                        const float* __restrict__ M0w, const float* __restrict__ M0b,
                        const float* __restrict__ M1w, const float* __restrict__ M1b,
                        const float* __restrict__ M2w, const float* __restrict__ M2b,
                        float* __restrict__ out, float invN) {
  if (threadIdx.x != 0 || blockIdx.x != 0) return;
  float hg[32], y1[16], y2[8];
  for (int j = 0; j < 32; ++j) hg[j] = sums[j] * invN;
  for (int j = 0; j < 16; ++j) {
    float a = M0b[j];
    for (int k = 0; k < 32; ++k) a += hg[k] * M0w[k * 16 + j];
    y1[j] = a > 0.f ? a : 0.f;
  }
  for (int j = 0; j < 8; ++j) {
    float a = M1b[j];
    for (int k = 0; k < 16; ++k) a += y1[k] * M1w[k * 8 + j];
    y2[j] = a > 0.f ? a : 0.f;
  }
  for (int j = 0; j < 10; ++j) {
    float a = M2b[j];
    for (int k = 0; k < 8; ++k) a += y2[k] * M2w[k * 10 + j];
    out[j] = a;
  }
}

extern "C" void kernel_launch(void* const* d_in, const int* in_sizes, int n_in,
                              void* d_out, int out_size, void* d_ws, size_t ws_size,
                              hipStream_t stream) {
  (void)in_sizes; (void)n_in; (void)out_size; (void)ws_size;
  const float* h      = (const float*)d_in[0];
  const int*   src    = (const int*)  d_in[1];
  const int*   dst    = (const int*)  d_in[2];
  const float* Wemb   = (const float*)d_in[3];
  const float* bemb   = (const float*)d_in[4];
  const float* Wp1    = (const float*)d_in[5];
  const float* bp1    = (const float*)d_in[6];
  const float* Wp2    = (const float*)d_in[7];
  const float* bp2    = (const float*)d_in[8];
  const float* convw  = (const float*)d_in[9];
  const float* convb  = (const float*)d_in[10];
  const float* Wep2   = (const float*)d_in[11];
  const float* bep2   = (const float*)d_in[12];
  const float* Wep3   = (const float*)d_in[13];
  const float* bep3   = (const float*)d_in[14];
  const float* bnlrg  = (const float*)d_in[15];
  const float* bnlrb  = (const float*)d_in[16];
  const float* Aw     = (const float*)d_in[17];
  const float* Ab     = (const float*)d_in[18];
  const float* Bw     = (const float*)d_in[19];
  const float* Bb     = (const float*)d_in[20];
  const float* Cw     = (const float*)d_in[21];
  const float* Cb     = (const float*)d_in[22];
  const float* Dw     = (const float*)d_in[23];
  const float* Db     = (const float*)d_in[24];
  const float* Ew     = (const float*)d_in[25];
  const float* Eb     = (const float*)d_in[26];
  const float* bnhg   = (const float*)d_in[27];
  const float* bnhb   = (const float*)d_in[28];
  const float* bneg   = (const float*)d_in[29];
  const float* bneb   = (const float*)d_in[30];
  const float* M0w    = (const float*)d_in[31];
  const float* M0b    = (const float*)d_in[32];
  const float* M1w    = (const float*)d_in[33];
  const float* M1b    = (const float*)d_in[34];
  const float* M2w    = (const float*)d_in[35];
  const float* M2b    = (const float*)d_in[36];

  float* ws = (float*)d_ws;
  size_t off = 0;
  float* Z     = ws + off; off += (size_t)N_NODES * BIGC;   // node fused outputs
  float* hh    = ws + off; off += (size_t)N_NODES * 32;
  float* e     = ws + off; off += (size_t)M_EDGES * 32;
  float* epre  = ws + off; off += (size_t)M_EDGES * 32;
  float* ABDE  = ws + off; off += (size_t)N_NODES * 128;
  float* num   = ws + off; off += (size_t)N_NODES * 32;
  float* den   = ws + off; off += (size_t)N_NODES * 32;
  float* hpre  = ws + off; off += (size_t)N_NODES * 32;
  float* Wbig  = ws + off; off += (size_t)IN_DIM * BIGC;
  float* bbig  = ws + off; off += BIGC;
  float* Wcat  = ws + off; off += 32 * 128;
  float* bcat  = ws + off; off += 128;
  float* stats = ws + off; off += 128;

  // ---- Stage A: fused node GEMM [N,128] @ [128,1152] via f32 WMMA ----
  build_wbig<<<(IN_DIM * BIGC + 255) / 256, 256, 0, stream>>>(
      Wemb, bemb, Wp1, bp1, Wp2, bp2, convw, convb, Wep2, bep2, Wbig, bbig);
  {
    int rowT = N_NODES / 16, colT = BIGC / 16;           // 1250 x 72
    int nT = rowT * colT;                                // 90000 tiles
    wmma_gemm_f32<IN_DIM, BIGC, BIGC, IN_DIM>
        <<<(nT + 3) / 4, 128, 0, stream>>>(h, Wbig, bbig, Z, rowT, colT);
  }
  copy_hh<<<2500, 256, 0, stream>>>(Z, hh, (long)N_NODES * 32);

  // ---- Stage B: edge low-rank score, BN, relu -> e ----
  edge_score<<<(M_EDGES + 7) / 8, 256, 0, stream>>>(Z, src, dst, Wep3, bep3, epre, M_EDGES);
  zero_f<<<1, 128, 0, stream>>>(stats, 128);
  colstats<<<512, 256, 0, stream>>>(epre, M_EDGES, stats);
  bn_finalize<<<1, 32, 0, stream>>>(stats, 1.0f / M_EDGES);
  bn_relu_apply<<<4096, 256, 0, stream>>>(epre, stats, bnlrg, bnlrb, e,
                                          (long)M_EDGES * 32, 0);

  // ---- GatedGCN layers ----
  for (int l = 0; l < L_LAYERS; ++l) {
    build_wcat<<<16, 256, 0, stream>>>(Aw, Bw, Dw, Ew, Ab, Bb, Db, Eb, l, Wcat, bcat);
    // [Ah|Bh|Dh|Eh] = hh @ Wcat : [N,32] x [32,128]
    wmma_gemm_f32<32, 128, 128, 32>
        <<<(1250 * 8) / 4, 128, 0, stream>>>(hh, Wcat, bcat, ABDE, 1250, 8);
    // Ce -> epre : [M,32] x [32,32]
    wmma_gemm_f32<32, 32, 32, 32>
        <<<(10000 * 2) / 4, 128, 0, stream>>>(e, Cw + l * 1024, Cb + l * 32,
                                              epre, 10000, 2);
    zero_f<<<640, 256, 0, stream>>>(num, (long)N_NODES * 32);
    zero_f<<<640, 256, 0, stream>>>(den, (long)N_NODES * 32);
    edge_update<<<4096, 256, 0, stream>>>(epre, ABDE, src, dst, num, den,
                                          (long)M_EDGES * 32);
    node_combine<<<1280, 256, 0, stream>>>(ABDE, num, den, hpre, (long)N_NODES * 32);
    // BN(h) + relu + residual
    zero_f<<<1, 128, 0, stream>>>(stats, 128);
    colstats<<<512, 256, 0, stream>>>(hpre, N_NODES, stats);
    bn_finalize<<<1, 32, 0, stream>>>(stats, 1.0f / N_NODES);
    bn_relu_apply<<<1280, 256, 0, stream>>>(hpre, stats, bnhg + l * 32, bnhb + l * 32,
                                            hh, (long)N_NODES * 32, 1);
    // BN(e) + relu + residual
    zero_f<<<1, 128, 0, stream>>>(stats, 128);
    colstats<<<512, 256, 0, stream>>>(epre, M_EDGES, stats);
    bn_finalize<<<1, 32, 0, stream>>>(stats, 1.0f / M_EDGES);
    bn_relu_apply<<<4096, 256, 0, stream>>>(epre, stats, bneg + l * 32, bneb + l * 32,
                                            e, (long)M_EDGES * 32, 1);
  }

  // ---- readout ----
  zero_f<<<1, 128, 0, stream>>>(stats, 128);
  colstats<<<512, 256, 0, stream>>>(hh, N_NODES, stats);
  readout<<<1, 32, 0, stream>>>(stats, M0w, M0b, M1w, M1b, M2w, M2b,
                                (float*)d_out, 1.0f / N_NODES);
}